// ImageCaptioningCNN_82171314307511
// MI455X (gfx1250) — compile-verified
//
#include <hip/hip_runtime.h>

// ============================================================================
// Image-captioning decode, MI455X (gfx1250, wave32).
//
// Key insight: the reference's VGG16 output is DEAD (only fixes batch size),
// and the O(L^2) LSTM recompute collapses to an O(L) incremental LSTM.
// Per step: two small fp32 GEMMs + argmax + gather, via v_wmma_f32_16x16x4_f32
// (fp32 keeps reference numerics so greedy argmax matches; weights are
// L2-resident at 36 MB << 192 MB so the chain is latency bound).
//
// Round-2: fused the LSTM gate nonlinearity into the gate GEMM — each wave
// owns cols {n0, 512+n0, 1024+n0, 1536+n0} (i,f,g,o accumulators), applies
// sigmoid/tanh on its C tiles and writes h,c directly. Eliminates the z
// round-trip and one kernel per step (3 launches/step). h is ping-ponged
// across steps (all blocks read h while owners write it); c is element-
// exclusive so updates in place.
//
// Shapes: B=8, E=256, U=512, V=10000, T=50.
// ============================================================================

typedef __attribute__((ext_vector_type(2))) float v2f;
typedef __attribute__((ext_vector_type(8))) float v8f;

#define B    8
#define E    256
#define U    512
#define V    10000
#define TLEN 50
#define G4U  2048        // 4*U

#define KA   768         // x|h concat K for gate GEMM
#define LDA  772         // padded LDS row stride (772 % 64 == 4 -> conflict-free b64)
#define LDH  516         // padded LDS row stride for h panel (516 % 64 == 4)

// --- WMMA f32 16x16x4 --------------------------------------------------------
// A (16x4): lanes 0-15 rows 0..15; VGPR0={K0|K2}, VGPR1={K1|K3} -> per-lane
//   contiguous pair at (k + 2*half). B (4x16): VGPR0={row0|row1}, VGPR1={row2|row3}.
// C/D (16x16): VGPR j: lanes 0-15 = row j, lanes 16-31 = row j+8.
__device__ __forceinline__ v8f wmma4(v8f c, v2f a, float b0, float b1) {
    v2f b; b.x = b0; b.y = b1;
    return __builtin_amdgcn_wmma_f32_16x16x4_f32(false, a, false, b,
                                                 (short)0, c, false, false);
}

__device__ __forceinline__ float sigf(float v) { return 1.0f / (1.0f + expf(-v)); }

// --- monotone float -> u32 key (order-preserving), packed with ~index --------
__device__ __forceinline__ unsigned long long pack_key(float v, unsigned idx) {
    unsigned vb = __float_as_uint(v);
    vb = (vb & 0x80000000u) ? ~vb : (vb | 0x80000000u);
    return ((unsigned long long)vb << 32) | (unsigned long long)(~idx);
}

// ============================================================================
// K0: init. h0=c=0, x = emb[token 0], cap[:,0,:] = emb[0], packed = 0.
// ============================================================================
__global__ __launch_bounds__(256) void k_init(const float* __restrict__ emb,
                                              float* __restrict__ x,
                                              float* __restrict__ h0,
                                              float* __restrict__ c,
                                              unsigned long long* __restrict__ packed,
                                              float* __restrict__ cap) {
    int tid = blockIdx.x * blockDim.x + threadIdx.x;      // 0..4095
    h0[tid] = 0.0f;                                       // B*U = 4096
    c[tid]  = 0.0f;
    if (tid < B * E) {                                    // 2048
        int r = tid >> 8, e = tid & (E - 1);
        float v = emb[e];                                 // token 0 row
        x[tid] = v;
        cap[(r * TLEN + 0) * E + e] = v;
    }
    if (tid < B) packed[tid] = 0ull;
}

// ============================================================================
// K1: fused LSTM step. For unit cols n0..n0+15, accumulate the four gate
// pre-activations (i,f,g,o) = [x|hin] @ [Wk;Wr] cols {n0, 512+n0, 1024+n0,
// 1536+n0}, add bl, apply gates, write c (in place) and hout (ping-pong).
// 8 blocks x 128 threads (4 waves): 32 waves cover 512 unit cols.
// A panel (16x768, rows 8-15 = 0) staged in LDS per block -> branch-free loop.
// ============================================================================
__global__ __launch_bounds__(128) void k_lstm_step(const float* __restrict__ x,
                                                   const float* __restrict__ hin,
                                                   float* __restrict__ hout,
                                                   float* __restrict__ c,
                                                   const float* __restrict__ Wk,
                                                   const float* __restrict__ Wr,
                                                   const float* __restrict__ bl) {
    __shared__ float lds_a[16 * LDA];

    for (int i = threadIdx.x; i < 16 * KA; i += 128) {
        int row = i / KA, col = i - row * KA;
        float v = 0.0f;
        if (row < 8) v = (col < E) ? x[row * E + col] : hin[row * U + (col - E)];
        lds_a[row * LDA + col] = v;
    }
    __syncthreads();

    int wave = threadIdx.x >> 5;                    // 0..3
    int lane = threadIdx.x & 31;
    int half = lane >> 4, m = lane & 15;
    int n0 = (blockIdx.x * 4 + wave) * 16;          // unit-col base, 0..496
    const float* arow = &lds_a[m * LDA];

    v8f ci = {}, cf = {}, cg = {}, co = {};
#pragma unroll 4
    for (int k = 0; k < E; k += 4) {                // x @ Wk, K=256
        v2f a = *(const v2f*)(arow + k + 2 * half);
        const float* w0 = Wk + (k + half) * G4U + n0 + m;      // row k|k+1
        const float* w1 = Wk + (k + 2 + half) * G4U + n0 + m;  // row k+2|k+3
        ci = wmma4(ci, a, w0[0],     w1[0]);        // gate i: cols n0
        cf = wmma4(cf, a, w0[U],     w1[U]);        // gate f: cols 512+n0
        cg = wmma4(cg, a, w0[2 * U], w1[2 * U]);    // gate g: cols 1024+n0
        co = wmma4(co, a, w0[3 * U], w1[3 * U]);    // gate o: cols 1536+n0
    }
#pragma unroll 4
    for (int k = 0; k < U; k += 4) {                // hin @ Wr, K=512
        v2f a = *(const v2f*)(arow + E + k + 2 * half);
        const float* w0 = Wr + (k + half) * G4U + n0 + m;
        const float* w1 = Wr + (k + 2 + half) * G4U + n0 + m;
        ci = wmma4(ci, a, w0[0],     w1[0]);
        cf = wmma4(cf, a, w0[U],     w1[U]);
        cg = wmma4(cg, a, w0[2 * U], w1[2 * U]);
        co = wmma4(co, a, w0[3 * U], w1[3 * U]);
    }

    if (half == 0) {                                // rows 0..7 live in lanes 0-15
        float bi = bl[n0 + m],         bf = bl[U + n0 + m];
        float bg = bl[2 * U + n0 + m], bo = bl[3 * U + n0 + m];
#pragma unroll
        for (int j = 0; j < 8; ++j) {
            int idx = j * U + n0 + m;
            float cn = sigf(cf[j] + bf) * c[idx] + sigf(ci[j] + bi) * tanhf(cg[j] + bg);
            c[idx]    = cn;
            hout[idx] = sigf(co[j] + bo) * tanhf(cn);
        }
    }
}

// ============================================================================
// K2: logits = h @ Wd + bd ([8x512]@[512x10000]); fused greedy argmax.
// 40 blocks x 512 threads (16 waves) -> 640 waves; 625 N-tiles of 16 cols.
// h panel staged in LDS; tail waves join fill + barrier then skip compute
// wave-uniformly (EXEC full for WMMA). Per row: wave shuffle max, then
// GLOBAL_ATOMIC_MAX_U64 on packed (val, ~idx) -> ties pick lowest index.
// ============================================================================
__global__ __launch_bounds__(512) void k_logits_argmax(const float* __restrict__ h,
                                                       const float* __restrict__ Wd,
                                                       const float* __restrict__ bd,
                                                       unsigned long long* __restrict__ packed) {
    __shared__ float lds_h[16 * LDH];

    for (int i = threadIdx.x; i < 16 * U; i += 512) {
        int row = i >> 9, col = i & (U - 1);
        lds_h[row * LDH + col] = (row < 8) ? h[row * U + col] : 0.0f;
    }
    __syncthreads();

    int wave = threadIdx.x >> 5;
    int tile = blockIdx.x * 16 + wave;
    if (tile < V / 16) {                            // wave-uniform: EXEC stays full
        int lane = threadIdx.x & 31;
        int half = lane >> 4, m = lane & 15;
        int n0 = tile * 16;
        const float* arow = &lds_h[m * LDH];

        v8f acc = {};
#pragma unroll 4
        for (int k = 0; k < U; k += 4) {
            v2f a = *(const v2f*)(arow + k + 2 * half);
            const float* w0 = Wd + (k + half) * V + n0 + m;
            const float* w1 = Wd + (k + 2 + half) * V + n0 + m;
            acc = wmma4(acc, a, w0[0], w1[0]);
        }

        float bias = bd[n0 + m];
#pragma unroll
        for (int j = 0; j < 8; ++j) {               // rows 0..7 (lanes 0-15 real)
            unsigned long long key =
                (half == 0) ? pack_key(acc[j] + bias, (unsigned)(n0 + m)) : 0ull;
#pragma unroll
            for (int off = 16; off > 0; off >>= 1) {
                unsigned long long o = __shfl_xor(key, off);
                key = (o > key) ? o : key;
            }
            if (lane == 0) atomicMax(&packed[j], key);
        }
    }
}

// ============================================================================
// K3: decode token from packed key, x = emb[tok], cap[:,t,:] = x, reset packed.
// 8 blocks (one per batch row) x 256 threads; block-local reset after barrier.
// ============================================================================
__global__ __launch_bounds__(256) void k_embed(unsigned long long* __restrict__ packed,
                                               const float* __restrict__ emb,
                                               float* __restrict__ x,
                                               float* __restrict__ cap,
                                               int t) {
    int r = blockIdx.x, e = threadIdx.x;
    unsigned long long key = packed[r];
    unsigned tok = ~((unsigned)(key & 0xFFFFFFFFull));
    float v = emb[(size_t)tok * E + e];
    x[r * E + e] = v;
    cap[(r * TLEN + t) * E + e] = v;
    __syncthreads();
    if (e == 0) packed[r] = 0ull;                  // ready for next step's atomicMax
}

// ============================================================================
// Host side: 1 init + 49 x 3 kernels, all on `stream` (graph-capturable).
// Inputs indexed from the END of d_in so vgg_params flattening is irrelevant
// (VGG is dead code): ... emb, Wk, Wr, bl, Wd, bd.
// ============================================================================
extern "C" void kernel_launch(void* const* d_in, const int* in_sizes, int n_in,
                              void* d_out, int out_size, void* d_ws, size_t ws_size,
                              hipStream_t stream) {
    const float* emb = (const float*)d_in[n_in - 6];   // [V,E]
    const float* Wk  = (const float*)d_in[n_in - 5];   // [E,4U]
    const float* Wr  = (const float*)d_in[n_in - 4];   // [U,4U]
    const float* bl  = (const float*)d_in[n_in - 3];   // [4U]
    const float* Wd  = (const float*)d_in[n_in - 2];   // [U,V]
    const float* bd  = (const float*)d_in[n_in - 1];   // [V]

    float* ws = (float*)d_ws;
    float* x  = ws;                       // [B,E]   2048 f
    float* h0 = ws + 2048;                // [B,U]   4096 f  (ping)
    float* h1 = ws + 6144;                // [B,U]   4096 f  (pong)
    float* c  = ws + 10240;               // [B,U]   4096 f
    unsigned long long* packed = (unsigned long long*)(ws + 14336);  // [B], 8B aligned
    float* cap = (float*)d_out;           // [B,TLEN,E]

    k_init<<<16, 256, 0, stream>>>(emb, x, h0, c, packed, cap);
    for (int t = 1; t < TLEN; ++t) {
        const float* hin = (t & 1) ? h0 : h1;      // t=1 reads zeroed h0
        float*       hout = (t & 1) ? h1 : h0;
        k_lstm_step<<<8, 128, 0, stream>>>(x, hin, hout, c, Wk, Wr, bl);
        k_logits_argmax<<<40, 512, 0, stream>>>(hout, Wd, bd, packed);
        k_embed<<<B, 256, 0, stream>>>(packed, emb, x, cap, t);
    }
}